// AI4DEM_33749853012358
// MI455X (gfx1250) — compile-verified
//
#include <hip/hip_runtime.h>

// ---------------------------------------------------------------------------
// AI4DEM contact forces on MI455X (gfx1250).
// L2-latency-bound irregular gather: 12.5M probes into a 4MB L2-resident
// cell->particle table. CDNA5 strategy: GLOBAL_LOAD_ASYNC_TO_LDS_B32 gather
// pipeline (ASYNCcnt), 4-deep per-wave ring buffer in LDS, issue-ahead-3;
// float4-packed particle data so each contact is 2x b128 gathers.
// ---------------------------------------------------------------------------

#define DGRID   100
#define NCELLS  (DGRID * DGRID * DGRID)
#define RADIUS  2                 // filter_size = 5 -> offsets in [-2, 2]
#define KDIM    (2 * RADIUS + 1)  // 5
#define NROWS   (KDIM * KDIM)     // 25 (z,y) rows of 5 x-probes
#define BLOCK   256
#define WAVES   (BLOCK / 32)
#define NBUF    4                 // async pipeline depth (rows in flight)
#define ROWINTS (32 * KDIM)       // ints per buffer slot (per wave)

#define WAITA(imm) asm volatile("s_wait_asynccnt " #imm ::: "memory")

__global__ void dem_init_grid(int* __restrict__ grid) {
    int i = blockIdx.x * blockDim.x + threadIdx.x;
    if (i < NCELLS) grid[i] = -1;
}

__global__ void dem_scatter(const float* __restrict__ x,
                            const float* __restrict__ y,
                            const float* __restrict__ z,
                            const float* __restrict__ pd,
                            int n, int* __restrict__ grid) {
    int i = blockIdx.x * blockDim.x + threadIdx.x;
    if (i >= n) return;
    const float inv_d = 1.0f / *pd;
    int cx = (int)rintf(x[i] * inv_d);
    int cy = (int)rintf(y[i] * inv_d);
    int cz = (int)rintf(z[i] * inv_d);
    if ((unsigned)cx < DGRID && (unsigned)cy < DGRID && (unsigned)cz < DGRID)
        grid[(cz * DGRID + cy) * DGRID + cx] = i;
}

__global__ void dem_pack(const float* __restrict__ x,  const float* __restrict__ y,
                         const float* __restrict__ z,  const float* __restrict__ vx,
                         const float* __restrict__ vy, const float* __restrict__ vz,
                         int n, float4* __restrict__ pos4, float4* __restrict__ vel4) {
    int i = blockIdx.x * blockDim.x + threadIdx.x;
    if (i >= n) return;
    pos4[i] = make_float4(x[i], y[i], z[i], 0.0f);
    vel4[i] = make_float4(vx[i], vy[i], vz[i], 0.0f);
}

__device__ __forceinline__ int wrapD(int v) {
    if (v < 0)            v += DGRID;
    else if (v >= DGRID)  v -= DGRID;
    return v;
}

// Issue one (z,y) row: 5 per-lane async gathers of grid indices into LDS.
__device__ __forceinline__ void issue_row(const int* __restrict__ row_base,
                                          const int* xs, unsigned lds_off) {
#pragma unroll
    for (int k = 0; k < KDIM; ++k) {
        unsigned long long ga = (unsigned long long)(size_t)(row_base + xs[k]);
        unsigned la = lds_off + 4u * (unsigned)k;
        asm volatile("global_load_async_to_lds_b32 %0, %1, off"
                     :: "v"(la), "v"(ga) : "memory");
    }
}

// Consume one row of 5 neighbor ids from LDS and accumulate pair forces.
__device__ __forceinline__ void consume_row(
    const int* __restrict__ ldsrow, int i,
    const float4* __restrict__ pos4, const float4* __restrict__ vel4,
    float xi, float yi, float zi, float vxi, float vyi, float vzi,
    float twod, float kn, float eta,
    float& fxc, float& fyc, float& fzc,
    float& fxd, float& fyd, float& fzd)
{
#pragma unroll
    for (int k = 0; k < KDIM; ++k) {
        const int j = ldsrow[k];
        if (j >= 0 && j != i) {
            const float4 pj = pos4[j];           // one b128 gather
            const float dxp = xi - pj.x;
            const float dyp = yi - pj.y;
            const float dzp = zi - pj.z;
            const float dist = sqrtf(dxp * dxp + dyp * dyp + dzp * dzp);
            if (dist < twod) {
                const float denom  = fmaxf(1e-4f, dist);
                const float invden = 1.0f / denom;
                // spring (collision) force
                const float coef = kn * (dist - twod) * invden;
                fxc += coef * dxp;
                fyc += coef * dyp;
                fzc += coef * dzp;
                // normal damping force
                const float4 vj = vel4[j];       // one b128 gather
                const float dvx = vxi - vj.x;
                const float dvy = vyi - vj.y;
                const float dvz = vzi - vj.z;
                const float vn = (dvx * dxp + dvy * dyp + dvz * dzp) * invden;
                const float dc = eta * vn * invden;
                fxd += dc * dxp;
                fyd += dc * dyp;
                fzd += dc * dzp;
            }
        }
    }
}

__global__ __launch_bounds__(BLOCK) void dem_forces(
    const float4* __restrict__ pos4, const float4* __restrict__ vel4,
    const float* __restrict__ pd, const float* __restrict__ pkn,
    const float* __restrict__ peta,
    const int* __restrict__ grid, int n, float* __restrict__ out)
{
    __shared__ int nbr[WAVES][NBUF][ROWINTS];   // 20 KB: per-wave 4-slot ring

    const int tid  = blockIdx.x * blockDim.x + threadIdx.x;
    const int lane = threadIdx.x & 31;
    const int wave = threadIdx.x >> 5;

    const bool active = (tid < n);
    const int  i = active ? tid : 0;   // keep EXEC full; mask only the store

    const float d     = *pd;
    const float kn    = *pkn;
    const float eta   = *peta;
    const float twod  = 2.0f * d;
    const float inv_d = 1.0f / d;

    const float4 pi = pos4[i];
    const float4 vi = vel4[i];
    const float xi = pi.x, yi = pi.y, zi = pi.z;
    const float vxi = vi.x, vyi = vi.y, vzi = vi.z;

    const int cx = (int)rintf(xi * inv_d);
    const int cy = (int)rintf(yi * inv_d);
    const int cz = (int)rintf(zi * inv_d);

    int xs[KDIM], ys[KDIM], zs[KDIM];
#pragma unroll
    for (int k = 0; k < KDIM; ++k) {
        xs[k] = wrapD(cx + k - RADIUS);
        ys[k] = wrapD(cy + k - RADIUS);
        zs[k] = wrapD(cz + k - RADIUS);
    }

    // LDS addressing for this wave/lane
    int* wbase = &nbr[wave][0][0];
    const unsigned lds_base = (unsigned)(size_t)wbase;   // low 32b = DS offset
    const int      lane_off = lane * KDIM;

    float fxc = 0.0f, fyc = 0.0f, fzc = 0.0f;
    float fxd = 0.0f, fyd = 0.0f, fzd = 0.0f;

    // -- prologue: prime rows 0..2 into ring slots 0..2 --------------------
#pragma unroll
    for (int r = 0; r < NBUF - 1; ++r) {
        const int* rb = grid + (zs[r / KDIM] * DGRID + ys[r % KDIM]) * DGRID;
        issue_row(rb, xs, lds_base + 4u * (unsigned)(r * ROWINTS + lane_off));
    }

    // -- steady state: issue row r+3, wait row r done (<=15 in flight) -----
    for (int r = 0; r < NROWS - (NBUF - 1); ++r) {
        const int rn = r + (NBUF - 1);
        const int* rb = grid + (zs[rn / KDIM] * DGRID + ys[rn % KDIM]) * DGRID;
        issue_row(rb, xs,
                  lds_base + 4u * (unsigned)((rn & (NBUF - 1)) * ROWINTS + lane_off));
        WAITA(0xf);
        consume_row(wbase + (r & (NBUF - 1)) * ROWINTS + lane_off, i,
                    pos4, vel4, xi, yi, zi, vxi, vyi, vzi, twod, kn, eta,
                    fxc, fyc, fzc, fxd, fyd, fzd);
    }

    // -- epilogue: rows 22, 23, 24 (ring slots 2, 3, 0) --------------------
    WAITA(0xa);
    consume_row(wbase + ((NROWS - 3) & (NBUF - 1)) * ROWINTS + lane_off, i,
                pos4, vel4, xi, yi, zi, vxi, vyi, vzi, twod, kn, eta,
                fxc, fyc, fzc, fxd, fyd, fzd);
    WAITA(0x5);
    consume_row(wbase + ((NROWS - 2) & (NBUF - 1)) * ROWINTS + lane_off, i,
                pos4, vel4, xi, yi, zi, vxi, vyi, vzi, twod, kn, eta,
                fxc, fyc, fzc, fxd, fyd, fzd);
    WAITA(0x0);
    consume_row(wbase + ((NROWS - 1) & (NBUF - 1)) * ROWINTS + lane_off, i,
                pos4, vel4, xi, yi, zi, vxi, vyi, vzi, twod, kn, eta,
                fxc, fyc, fzc, fxd, fyd, fzd);

    if (active) {
        out[0 * n + tid] = fxc;
        out[1 * n + tid] = fyc;
        out[2 * n + tid] = fzc;
        out[3 * n + tid] = fxd;
        out[4 * n + tid] = fyd;
        out[5 * n + tid] = fzd;
    }
}

extern "C" void kernel_launch(void* const* d_in, const int* in_sizes, int n_in,
                              void* d_out, int out_size, void* d_ws, size_t ws_size,
                              hipStream_t stream) {
    const float* x   = (const float*)d_in[0];
    const float* y   = (const float*)d_in[1];
    const float* z   = (const float*)d_in[2];
    const float* vx  = (const float*)d_in[3];
    const float* vy  = (const float*)d_in[4];
    const float* vz  = (const float*)d_in[5];
    const float* pd  = (const float*)d_in[6];
    const float* pkn = (const float*)d_in[7];
    const float* pet = (const float*)d_in[8];
    // d_in[9] friction, d_in[10] dt: unused by the reference output
    const int n = in_sizes[0];

    // scratch layout: [grid: 4MB][pos4: 16n][vel4: 16n]  (all 16B aligned)
    char* ws = (char*)d_ws;
    int*    grid = (int*)ws;
    float4* pos4 = (float4*)(ws + (size_t)NCELLS * sizeof(int));
    float4* vel4 = pos4 + n;
    float*  out  = (float*)d_out;

    dem_init_grid<<<(NCELLS + 255) / 256, 256, 0, stream>>>(grid);
    dem_scatter<<<(n + 255) / 256, 256, 0, stream>>>(x, y, z, pd, n, grid);
    dem_pack<<<(n + 255) / 256, 256, 0, stream>>>(x, y, z, vx, vy, vz, n, pos4, vel4);
    dem_forces<<<(n + BLOCK - 1) / BLOCK, BLOCK, 0, stream>>>(
        pos4, vel4, pd, pkn, pet, grid, n, out);
}